// GarNet_85856396247497
// MI455X (gfx1250) — compile-verified
//
#include <hip/hip_runtime.h>

// GarNet — factored formulation for MI455X (gfx1250, wave32, WMMA).
//
// out[b,v,f] = mask * ( sum_a ew[b,v,a] * M[b,a,f] + b_out[f] )
//   M[b,a,f] = sum_p agg[b,a,p] * w_out[a*32+p, f]
//   agg[b,a,p] = (1/V) sum_v ew[b,v,a] * feat[b,v,p]
//   ew = mask * 2^(-(data@w_s + b_s)^2),  feat = data@w_flr + b_flr

#define B_     64
#define V_     2048
#define FIN_   16
#define NAGG_  8
#define NPROP_ 32
#define NFILT_ 64

typedef __attribute__((ext_vector_type(2))) float v2f;
typedef __attribute__((ext_vector_type(8))) float v8f;

// ---------------------------------------------------------------------------
// Kernel 1: per-batch aggregation.  grid = B, block = 256 (8 wave32s).
// Produces ew[b][V][8] (global scratch) and M[b][8][64] (global scratch).
// ---------------------------------------------------------------------------
__global__ __launch_bounds__(256) void garnet_agg_kernel(
    const float* __restrict__ data,        // [B][V][16]
    const int*   __restrict__ num_vertex,  // [B][1]
    const float* __restrict__ w_flr,       // [16][32]
    const float* __restrict__ b_flr,       // [32]
    const float* __restrict__ w_s,         // [16][8]
    const float* __restrict__ b_s,         // [8]
    const float* __restrict__ w_out,       // [256][64]
    float*       __restrict__ ew_out,      // [B][V][8]
    float*       __restrict__ m_out)       // [B][8][64]
{
    __shared__ float s_ws[FIN_ * NAGG_];    // 128
    __shared__ float s_wf[FIN_ * NPROP_];   // 512
    __shared__ float s_bs[NAGG_];
    __shared__ float s_bf[NPROP_];
    __shared__ float s_ew[256 * 9];         // chunk edge weights, padded stride 9
    __shared__ float s_ft[256 * 33];        // chunk features, padded stride 33
    __shared__ float s_agg[NAGG_ * NPROP_]; // 256

    const int b   = blockIdx.x;
    const int tid = threadIdx.x;
    const int nv  = num_vertex[b];

    if (tid < FIN_ * NAGG_) s_ws[tid] = w_s[tid];
    s_wf[tid]       = w_flr[tid];
    s_wf[tid + 256] = w_flr[tid + 256];
    if (tid < NAGG_)  s_bs[tid] = b_s[tid];
    if (tid < NPROP_) s_bf[tid] = b_flr[tid];
    __syncthreads();

    const int ap_a = tid >> 5;  // wave id -> aggregator index (0..7)
    const int ap_p = tid & 31;  // lane    -> propagator index (0..31)
    float acc = 0.0f;

    for (int chunk = 0; chunk < V_; chunk += 256) {
        const int v = chunk + tid;

        float x[FIN_];
        const float* dp = data + ((size_t)b * V_ + v) * FIN_;
#pragma unroll
        for (int i = 0; i < FIN_; ++i) x[i] = dp[i];

        const float msk = (v < nv) ? 1.0f : 0.0f;
        float* ewg = ew_out + ((size_t)b * V_ + v) * NAGG_;
#pragma unroll
        for (int a = 0; a < NAGG_; ++a) {
            float d = s_bs[a];
#pragma unroll
            for (int i = 0; i < FIN_; ++i) d += x[i] * s_ws[i * NAGG_ + a];
            // exp(-ln2*d^2) == 2^(-d^2)  ->  single v_exp_f32
            const float e = msk * __builtin_amdgcn_exp2f(-d * d);
            ewg[a]            = e;
            s_ew[tid * 9 + a] = e;
        }
#pragma unroll
        for (int p = 0; p < NPROP_; ++p) {
            float ft = s_bf[p];
#pragma unroll
            for (int i = 0; i < FIN_; ++i) ft += x[i] * s_wf[i * NPROP_ + p];
            s_ft[tid * 33 + p] = ft;
        }
        __syncthreads();

        // Each thread owns one (a,p) pair; reduce this 256-vertex chunk.
        // s_ew read is a lane-broadcast, s_ft read is bank-conflict-free.
        for (int vv = 0; vv < 256; ++vv)
            acc += s_ew[vv * 9 + ap_a] * s_ft[vv * 33 + ap_p];
        __syncthreads();
    }

    s_agg[ap_a * NPROP_ + ap_p] = acc * (1.0f / (float)V_);
    __syncthreads();

    // Fold w_out: M[a][f] = sum_p agg[a][p] * w_out[a*32+p][f]
#pragma unroll
    for (int q = tid; q < NAGG_ * NFILT_; q += 256) {
        const int a = q / NFILT_;
        const int f = q % NFILT_;
        float m = 0.0f;
#pragma unroll
        for (int p = 0; p < NPROP_; ++p)
            m += s_agg[a * NPROP_ + p] * w_out[(a * NPROP_ + p) * NFILT_ + f];
        m_out[(size_t)b * (NAGG_ * NFILT_) + q] = m;
    }
}

// ---------------------------------------------------------------------------
// Kernel 2: out[b,v,f] = mask * (E(Vx8) @ M(8x64) + bias) via fp32 WMMA.
// One wave per 16x16 output tile; K=8 -> two V_WMMA_F32_16X16X4_F32.
// Fragment layouts per CDNA5 ISA 7.12.2:
//   A 16x4:  lanes 0-15 {v0=K0, v1=K1}; lanes 16-31 {v0=K2, v1=K3}
//   B 4x16:  v0 = rows {K0 | K2} across lane halves, v1 = rows {K1 | K3}
//   C/D 16x16: VGPR i holds rows i (lanes 0-15) and i+8 (lanes 16-31)
// ---------------------------------------------------------------------------
__global__ __launch_bounds__(256) void garnet_out_kernel(
    const float* __restrict__ ew,          // [B][V][8]
    const float* __restrict__ mmat,        // [B][8][64]
    const float* __restrict__ b_out,       // [64]
    const int*   __restrict__ num_vertex,  // [B][1]
    float*       __restrict__ out)         // [B][V][64]
{
    const int lane = threadIdx.x & 31;
    const int wave = threadIdx.x >> 5;
    const int tile = blockIdx.x * 8 + wave;     // 512 tiles per batch
    const int b     = tile >> 9;
    const int rem   = tile & 511;
    const int vbase = (rem >> 2) * 16;
    const int fbase = (rem & 3) * 16;

    const int half = lane >> 4;
    const int lr   = lane & 15;
    const int f    = fbase + lr;

    // A fragments: rows = 16 vertices, K = aggregator index.
    const float* ewrow = ew + ((size_t)b * V_ + (vbase + lr)) * NAGG_;
    v2f a0, a1;
    a0.x = ewrow[2 * half + 0];
    a0.y = ewrow[2 * half + 1];
    a1.x = ewrow[2 * half + 4];
    a1.y = ewrow[2 * half + 5];

    // B fragments: rows = aggregator index, cols = 16 filters.
    const float* mb = mmat + (size_t)b * (NAGG_ * NFILT_);
    v2f b0, b1;
    b0.x = mb[(2 * half + 0) * NFILT_ + f];
    b0.y = mb[(2 * half + 1) * NFILT_ + f];
    b1.x = mb[(2 * half + 4) * NFILT_ + f];
    b1.y = mb[(2 * half + 5) * NFILT_ + f];

    // C init: bias broadcast to all 16 rows of the tile.
    const float bias = b_out[f];
    v8f c;
#pragma unroll
    for (int i = 0; i < 8; ++i) c[i] = bias;

    // D = A0*B0 + A1*B1 + C   (K = 4..7 then 0..3)
    c = __builtin_amdgcn_wmma_f32_16x16x4_f32(false, a1, false, b1,
                                              (short)0, c, false, false);
    c = __builtin_amdgcn_wmma_f32_16x16x4_f32(false, a0, false, b0,
                                              (short)0, c, false, false);

    const int nv = num_vertex[b];
    float* outb = out + (size_t)b * V_ * NFILT_;
#pragma unroll
    for (int i = 0; i < 8; ++i) {
        const int v   = vbase + i + 8 * half;
        const float r = (v < nv) ? c[i] : 0.0f;   // zero bias on padded rows
        outb[(size_t)v * NFILT_ + f] = r;
    }
}

// ---------------------------------------------------------------------------
extern "C" void kernel_launch(void* const* d_in, const int* in_sizes, int n_in,
                              void* d_out, int out_size, void* d_ws, size_t ws_size,
                              hipStream_t stream) {
    const float* data       = (const float*)d_in[0];
    const int*   num_vertex = (const int*)  d_in[1];
    const float* w_flr      = (const float*)d_in[2];
    const float* b_flr      = (const float*)d_in[3];
    const float* w_s        = (const float*)d_in[4];
    const float* b_s        = (const float*)d_in[5];
    const float* w_out      = (const float*)d_in[6];
    const float* b_out      = (const float*)d_in[7];
    float*       out        = (float*)d_out;

    float* ws   = (float*)d_ws;
    float* ewb  = ws;                                  // B*V*8   floats (4 MB)
    float* mmat = ws + (size_t)B_ * V_ * NAGG_;        // B*8*64  floats (128 KB)

    garnet_agg_kernel<<<B_, 256, 0, stream>>>(
        data, num_vertex, w_flr, b_flr, w_s, b_s, w_out, ewb, mmat);

    const int tiles  = B_ * (V_ / 16) * (NFILT_ / 16); // 32768 wave-tiles
    const int blocks = tiles / 8;                      // 8 waves per block
    garnet_out_kernel<<<blocks, 256, 0, stream>>>(
        ewb, mmat, b_out, num_vertex, out);
}